// Sim2_16492674417473
// MI455X (gfx1250) — compile-verified
//
#include <hip/hip_runtime.h>
#include <hip/hip_bf16.h>

typedef __attribute__((ext_vector_type(2))) float v2f;
typedef __attribute__((ext_vector_type(8))) float v8f;

#define EPS 1e-3f

// ---------------------------------------------------------------------------
// Kernel A: cost volume (9x9, 3 past frames) + softmax(81) + 1x1 conv 243->4
// + BN + SiLU on the 20x20 level.  One wave (32 lanes) per output pixel.
// ---------------------------------------------------------------------------
__global__ void __launch_bounds__(256)
sim_kernel(const float* __restrict__ f2,       // [4,4,512,20,20]
           const float* __restrict__ w1,       // [4,243]
           const float* __restrict__ g1, const float* __restrict__ b1,
           const float* __restrict__ m1, const float* __restrict__ v1,
           float* __restrict__ sim_out)        // [4,4,20,20]
{
    __shared__ float s_lds[8][3 * 81];
    const int lane = threadIdx.x & 31;
    const int wave = threadIdx.x >> 5;
    const int p    = blockIdx.x * 8 + wave;    // 0..1599
    const int bb   = p / 400;
    const int hw   = p % 400;
    const int h    = hw / 20, w = hw % 20;

    // past frames t=0..2, channels c = lane + 32*k   (C=512 -> 16 per lane)
    float past[3][16];
    #pragma unroll
    for (int t = 0; t < 3; ++t)
        #pragma unroll
        for (int k = 0; k < 16; ++k) {
            const int c = lane + 32 * k;
            past[t][k] = f2[(((size_t)bb * 4 + t) * 512 + c) * 400 + hw];
        }
    const float* cur = f2 + ((size_t)bb * 4 + 3) * 512 * 400;

    for (int d = 0; d < 81; ++d) {
        const int dy = d / 9 - 4, dx = d % 9 - 4;
        const int hh = h + dy, ww = w + dx;
        const bool inb = (hh >= 0) & (hh < 20) & (ww >= 0) & (ww < 20);
        float s0 = 0.f, s1 = 0.f, s2 = 0.f;
        #pragma unroll
        for (int k = 0; k < 16; ++k) {
            const int c = lane + 32 * k;
            const float cv = inb ? cur[(size_t)c * 400 + hh * 20 + ww] : 0.0f;
            s0 += fabsf(cv - past[0][k]);
            s1 += fabsf(cv - past[1][k]);
            s2 += fabsf(cv - past[2][k]);
        }
        #pragma unroll
        for (int off = 16; off; off >>= 1) {      // wave32 reduction
            s0 += __shfl_xor(s0, off, 32);
            s1 += __shfl_xor(s1, off, 32);
            s2 += __shfl_xor(s2, off, 32);
        }
        if (lane == 0) {
            s_lds[wave][0 * 81 + d] = -s0;
            s_lds[wave][1 * 81 + d] = -s1;
            s_lds[wave][2 * 81 + d] = -s2;
        }
    }

    // softmax over the 81 displacements, per past frame.
    // lane owns d = lane, lane+32, and (if lane<17) lane+64.
    const int  d0 = lane, d1 = lane + 32, d2 = lane + 64;
    const bool has2 = (d2 < 81);
    for (int t = 0; t < 3; ++t) {
        float* s = &s_lds[wave][t * 81];
        float mx = fmaxf(s[d0], s[d1]);
        if (has2) mx = fmaxf(mx, s[d2]);
        #pragma unroll
        for (int off = 16; off; off >>= 1) mx = fmaxf(mx, __shfl_xor(mx, off, 32));
        const float e0 = __expf(s[d0] - mx);
        const float e1 = __expf(s[d1] - mx);
        const float e2 = has2 ? __expf(s[d2] - mx) : 0.0f;
        float sum = e0 + e1 + e2;
        #pragma unroll
        for (int off = 16; off; off >>= 1) sum += __shfl_xor(sum, off, 32);
        const float inv = 1.0f / sum;
        s[d0] = e0 * inv;
        s[d1] = e1 * inv;
        if (has2) s[d2] = e2 * inv;
    }

    // 1x1 conv 243 -> 4, lane-parallel over the 243 inputs
    float acc0 = 0.f, acc1 = 0.f, acc2 = 0.f, acc3 = 0.f;
    for (int j = lane; j < 243; j += 32) {
        const float pv = s_lds[wave][j];
        acc0 += w1[0 * 243 + j] * pv;
        acc1 += w1[1 * 243 + j] * pv;
        acc2 += w1[2 * 243 + j] * pv;
        acc3 += w1[3 * 243 + j] * pv;
    }
    #pragma unroll
    for (int off = 16; off; off >>= 1) {
        acc0 += __shfl_xor(acc0, off, 32);
        acc1 += __shfl_xor(acc1, off, 32);
        acc2 += __shfl_xor(acc2, off, 32);
        acc3 += __shfl_xor(acc3, off, 32);
    }
    if (lane == 0) {
        const float a[4] = {acc0, acc1, acc2, acc3};
        #pragma unroll
        for (int o = 0; o < 4; ++o) {
            float y = (a[o] - m1[o]) * (g1[o] * rsqrtf(v1[o] + EPS)) + b1[o];
            sim_out[((size_t)bb * 4 + o) * 400 + hw] = y / (1.0f + __expf(-y));
        }
    }
}

// ---------------------------------------------------------------------------
// Bilinear (align_corners=True) sample of sim [4,4,20,20] at a level pixel.
// HI/WI compile-time -> the scale folds to a constant (identity at 20x20).
// ---------------------------------------------------------------------------
template <int HI, int WI>
__device__ __forceinline__ float sim_sample(const float* __restrict__ sim,
                                            int b, int hc, int pix)
{
    const int hi = pix / WI, wi = pix % WI;
    const float fy = (float)hi * (19.0f / (float)(HI - 1));
    const float fx = (float)wi * (19.0f / (float)(WI - 1));
    const int y0 = (int)fy, x0 = (int)fx;
    const int y1 = min(y0 + 1, 19), x1 = min(x0 + 1, 19);
    const float wy = fy - (float)y0, wx = fx - (float)x0;
    const float* s = sim + ((size_t)b * 4 + hc) * 400;
    const float v00 = s[y0 * 20 + x0], v01 = s[y0 * 20 + x1];
    const float v10 = s[y1 * 20 + x0], v11 = s[y1 * 20 + x1];
    return v00 * (1.f - wy) * (1.f - wx) + v01 * (1.f - wy) * wx
         + v10 * wy * (1.f - wx)         + v11 * wy * wx;
}

// ---------------------------------------------------------------------------
// Kernel B: 1x1 conv (Ci+4 -> Ci) + BN + SiLU as a WMMA GEMM.
// One wave per block; each wave owns a 16(o) x 64(pixel) strip = 4 D tiles
// sharing the A (weight) fragment.  K marched in steps of 4 with
// V_WMMA_F32_16X16X4_F32; the 4 hidden (upsampled-sim) channels are exactly
// the peeled final K-chunk, so the hot loop is branch-free.
// Separate SSA pointers (no arrays) keep address-space inference intact so
// loads lower to global_load (not flat_load); CI/HI/WI compile-time turns
// the second-element offset (HW*4 bytes) into an instruction immediate.
//   A fragment (16x4 f32): lanes 0-15 hold K=kc,kc+1; lanes 16-31 K=kc+2,kc+3
//   B fragment (4x16 f32): mirrored (row striped across lanes)
//   C/D (16x16 f32): VGPR r -> M=r (lanes 0-15) / M=r+8 (lanes 16-31), N=lane%16
// ---------------------------------------------------------------------------
template <int CI, int HI, int WI>
__global__ void __launch_bounds__(32)
conv2_wmma_kernel(const float* __restrict__ feat,  // [4,4,CI,HI,WI]
                  const float* __restrict__ w2,    // [CI, CI+4]
                  const float* __restrict__ g2, const float* __restrict__ b2,
                  const float* __restrict__ m2, const float* __restrict__ v2,
                  const float* __restrict__ sim,   // [4,4,20,20]
                  float* __restrict__ out)         // [4,CI,HI,WI]
{
    constexpr int HW  = HI * WI;
    constexpr int CIN = CI + 4;
    const int lane = threadIdx.x;      // blockDim.x == 32
    const int half = lane >> 4;        // 0: lanes 0-15, 1: lanes 16-31
    const int lan  = lane & 15;
    const int mTile = blockIdx.y;
    const int tileStride = gridDim.x;  // nTiles / 4
    const int orow = mTile * 16 + lan; // weight row this lane feeds into A

    // 4 N-tiles strided across the pixel space (shared weights), kept in
    // individually named variables so everything stays in registers.
    const int nT0 = blockIdx.x;
    const int nT1 = blockIdx.x + tileStride;
    const int nT2 = blockIdx.x + 2 * tileStride;
    const int nT3 = blockIdx.x + 3 * tileStride;
    const int b0 = (nT0 * 16) / HW, pix0 = (nT0 * 16) % HW + lan;
    const int b1 = (nT1 * 16) / HW, pix1 = (nT1 * 16) % HW + lan;
    const int b2_ = (nT2 * 16) / HW, pix2 = (nT2 * 16) % HW + lan;
    const int b3 = (nT3 * 16) / HW, pix3 = (nT3 * 16) % HW + lan;

    const float* fp0 = feat + (((size_t)b0  * 4 + 3) * CI + 2 * half) * HW + pix0;
    const float* fp1 = feat + (((size_t)b1  * 4 + 3) * CI + 2 * half) * HW + pix1;
    const float* fp2 = feat + (((size_t)b2_ * 4 + 3) * CI + 2 * half) * HW + pix2;
    const float* fp3 = feat + (((size_t)b3  * 4 + 3) * CI + 2 * half) * HW + pix3;
    const float2* wp = (const float2*)(w2 + (size_t)orow * CIN + 2 * half);

    v8f acc0 = {}, acc1 = {}, acc2 = {}, acc3 = {};

    // Hot loop: global channels only (branch-free, EXEC always all-ones).
    #pragma unroll 2
    for (int kc = 0; kc < CI; kc += 4) {
        const float2 wv = *wp; wp += 2;        // advance 4 floats / chunk
        v2f a; a.x = wv.x; a.y = wv.y;
        v2f t0; t0.x = fp0[0]; t0.y = fp0[HW]; fp0 += 4 * HW;
        v2f t1; t1.x = fp1[0]; t1.y = fp1[HW]; fp1 += 4 * HW;
        v2f t2; t2.x = fp2[0]; t2.y = fp2[HW]; fp2 += 4 * HW;
        v2f t3; t3.x = fp3[0]; t3.y = fp3[HW]; fp3 += 4 * HW;
        acc0 = __builtin_amdgcn_wmma_f32_16x16x4_f32(false, a, false, t0, (short)0, acc0, false, false);
        acc1 = __builtin_amdgcn_wmma_f32_16x16x4_f32(false, a, false, t1, (short)0, acc1, false, false);
        acc2 = __builtin_amdgcn_wmma_f32_16x16x4_f32(false, a, false, t2, (short)0, acc2, false, false);
        acc3 = __builtin_amdgcn_wmma_f32_16x16x4_f32(false, a, false, t3, (short)0, acc3, false, false);
    }

    // Peeled final chunk: the 4 hidden channels (bilinear-upsampled sim).
    {
        const float2 wv = *wp;                 // weights at k = CI .. CI+3
        v2f a; a.x = wv.x; a.y = wv.y;
        const int hc0 = 2 * half, hc1 = 2 * half + 1;
        v2f t0; t0.x = sim_sample<HI, WI>(sim, b0,  hc0, pix0);
                t0.y = sim_sample<HI, WI>(sim, b0,  hc1, pix0);
        v2f t1; t1.x = sim_sample<HI, WI>(sim, b1,  hc0, pix1);
                t1.y = sim_sample<HI, WI>(sim, b1,  hc1, pix1);
        v2f t2; t2.x = sim_sample<HI, WI>(sim, b2_, hc0, pix2);
                t2.y = sim_sample<HI, WI>(sim, b2_, hc1, pix2);
        v2f t3; t3.x = sim_sample<HI, WI>(sim, b3,  hc0, pix3);
                t3.y = sim_sample<HI, WI>(sim, b3,  hc1, pix3);
        acc0 = __builtin_amdgcn_wmma_f32_16x16x4_f32(false, a, false, t0, (short)0, acc0, false, false);
        acc1 = __builtin_amdgcn_wmma_f32_16x16x4_f32(false, a, false, t1, (short)0, acc1, false, false);
        acc2 = __builtin_amdgcn_wmma_f32_16x16x4_f32(false, a, false, t2, (short)0, acc2, false, false);
        acc3 = __builtin_amdgcn_wmma_f32_16x16x4_f32(false, a, false, t3, (short)0, acc3, false, false);
    }

    // Fused BatchNorm(eval) + SiLU epilogue, coalesced stores over pix.
    #pragma unroll
    for (int r = 0; r < 8; ++r) {
        const int o = mTile * 16 + r + 8 * half;
        const float scale = g2[o] * rsqrtf(v2[o] + EPS);
        const float mean  = m2[o];
        const float bias  = b2[o];
        float y;
        y = (acc0[r] - mean) * scale + bias;
        out[((size_t)b0  * CI + o) * HW + pix0] = y / (1.0f + __expf(-y));
        y = (acc1[r] - mean) * scale + bias;
        out[((size_t)b1  * CI + o) * HW + pix1] = y / (1.0f + __expf(-y));
        y = (acc2[r] - mean) * scale + bias;
        out[((size_t)b2_ * CI + o) * HW + pix2] = y / (1.0f + __expf(-y));
        y = (acc3[r] - mean) * scale + bias;
        out[((size_t)b3  * CI + o) * HW + pix3] = y / (1.0f + __expf(-y));
    }
}

// ---------------------------------------------------------------------------
extern "C" void kernel_launch(void* const* d_in, const int* in_sizes, int n_in,
                              void* d_out, int out_size, void* d_ws, size_t ws_size,
                              hipStream_t stream)
{
    const float* f0 = (const float*)d_in[0];   // [4,4,128,80,80]
    const float* f1 = (const float*)d_in[1];   // [4,4,256,40,40]
    const float* f2 = (const float*)d_in[2];   // [4,4,512,20,20]
    // d_in[3], d_in[4]: clip ids (unused by the math)
    const float* w1  = (const float*)d_in[5];
    const float* g1  = (const float*)d_in[6];
    const float* b1  = (const float*)d_in[7];
    const float* m1  = (const float*)d_in[8];
    const float* v1  = (const float*)d_in[9];

    const float* w20 = (const float*)d_in[10];
    const float* g20 = (const float*)d_in[11];
    const float* b20 = (const float*)d_in[12];
    const float* m20 = (const float*)d_in[13];
    const float* v20 = (const float*)d_in[14];
    const float* w21 = (const float*)d_in[15];
    const float* g21 = (const float*)d_in[16];
    const float* b21 = (const float*)d_in[17];
    const float* m21 = (const float*)d_in[18];
    const float* v21 = (const float*)d_in[19];
    const float* w22 = (const float*)d_in[20];
    const float* g22 = (const float*)d_in[21];
    const float* b22 = (const float*)d_in[22];
    const float* m22 = (const float*)d_in[23];
    const float* v22 = (const float*)d_in[24];

    float* sim = (float*)d_ws;                 // [4,4,20,20] = 25.6 KB scratch
    float* out = (float*)d_out;

    // Stage 1: cost volume -> softmax -> conv1 -> sim  (1600 pixels, wave each)
    sim_kernel<<<200, 256, 0, stream>>>(f2, w1, g1, b1, m1, v1, sim);

    // Stage 2: per-level WMMA GEMM + BN + SiLU.  One wave per block,
    // grid = (nTiles/4, CI/16); each wave does 4 N-tiles.
    conv2_wmma_kernel<128, 80, 80><<<dim3(400, 8), 32, 0, stream>>>(
        f0, w20, g20, b20, m20, v20, sim, out);
    conv2_wmma_kernel<256, 40, 40><<<dim3(100, 16), 32, 0, stream>>>(
        f1, w21, g21, b21, m21, v21, sim, out + 3276800);
    conv2_wmma_kernel<512, 20, 20><<<dim3(25, 32), 32, 0, stream>>>(
        f2, w22, g22, b22, m22, v22, sim, out + 4915200);
}